// VectorQuantizerEMA_3582002725175
// MI455X (gfx1250) — compile-verified
//
#include <hip/hip_runtime.h>

// ---------------------------------------------------------------------------
// VQ-VAE vector quantizer for MI455X (gfx1250, wave32, WMMA).
// score(n,k) = 0.5*||e_k||^2 - x_n . e_k   (argmin-equivalent to L2 distance)
// Dot products via V_WMMA_F32_16X16X4_F32; codebook double-buffered through
// padded LDS using GLOBAL_LOAD_ASYNC_TO_LDS_B128 (ASYNCcnt) so DMA staging
// overlaps the WMMA stream.
// ---------------------------------------------------------------------------

constexpr int CDIM   = 128;    // embedding dim
constexpr int KCODES = 8192;   // num embeddings
constexpr int NROWS  = 8192;   // 8*32*32 flattened inputs
constexpr int NSPLIT = 8;      // code-range splits per row block (parallelism)
constexpr int ROWS_PER_WG = 128;   // 8 waves * 16 rows
constexpr int CHUNK  = 64;     // codes staged in LDS per iteration
constexpr int LDS_STRIDE = 132;    // 128 + 4 pad -> conflict-free ds_load_b64

typedef __attribute__((ext_vector_type(2))) float v2f;
typedef __attribute__((ext_vector_type(4))) float v4f;
typedef __attribute__((ext_vector_type(8))) float v8f;

// Async DMA: 16 bytes global -> LDS, per lane. Tracked by ASYNCcnt.
__device__ __forceinline__ void async_copy16(unsigned lds_off, const float* gaddr) {
  asm volatile("global_load_async_to_lds_b128 %0, %1, off"
               :: "v"(lds_off), "v"(gaddr)
               : "memory");
}
__device__ __forceinline__ void wait_async0() {
  asm volatile("s_wait_asynccnt 0x0" ::: "memory");
}

// ---- kernel 1: half squared norms of codebook rows -------------------------
__global__ __launch_bounds__(256) void halfnorm_kernel(const float* __restrict__ w,
                                                       float* __restrict__ hn) {
  const int wave = threadIdx.x >> 5, lane = threadIdx.x & 31;
  const int code = blockIdx.x * 8 + wave;
  const v4f v = *(const v4f*)(w + (size_t)code * CDIM + lane * 4);
  float s = v.x * v.x + v.y * v.y + v.z * v.z + v.w * v.w;
  for (int off = 16; off > 0; off >>= 1) s += __shfl_down(s, off, 32);
  if (lane == 0) hn[code] = 0.5f * s;
}

// ---- kernel 2: WMMA score tiles + running argmin ---------------------------
__global__ __launch_bounds__(256) void argmin_kernel(const float* __restrict__ x,
                                                     const float* __restrict__ w,
                                                     const float* __restrict__ hn,
                                                     float* __restrict__ pval,
                                                     int* __restrict__ pidx) {
  __shared__ float lds[2][CHUNK * LDS_STRIDE];   // double buffer, ~66 KB

  const int wave = threadIdx.x >> 5, lane = threadIdx.x & 31;
  const int row_block = blockIdx.x / NSPLIT;
  const int split     = blockIdx.x % NSPLIT;
  const int row0 = row_block * ROWS_PER_WG + wave * 16;
  const int l15  = lane & 15;
  const int hi2  = (lane >> 4) * 2;   // 0 for lanes 0-15 (K=0,1), 2 for 16-31 (K=2,3)

  // Per-thread staging coordinates: thread stages rows r0, r0+8, ..., r0+56
  // at column c4 (one b128 each) -> 256 threads cover 64 rows x 128 floats.
  const int r0 = threadIdx.x >> 5;          // 0..7
  const int c4 = (threadIdx.x & 31) * 4;    // 0..124 step 4
  const unsigned lds_base0 =
      (unsigned)(size_t)&lds[0][r0 * LDS_STRIDE + c4];
  const unsigned lds_base1 =
      (unsigned)(size_t)&lds[1][r0 * LDS_STRIDE + c4];

  // Pin this wave's 16x128 A-tile in VGPRs, laid out per ISA 16x4 f32 A-matrix:
  // lane<16: {K=c0,K=c0+1}; lane>=16: {K=c0+2,K=c0+3}, M = lane&15.
  v2f a[32];
#pragma unroll
  for (int kk = 0; kk < 32; ++kk)
    a[kk] = *(const v2f*)(x + (size_t)(row0 + l15) * CDIM + kk * 4 + hi2);

  float mval[8];
  int   midx[8];
#pragma unroll
  for (int s = 0; s < 8; ++s) { mval[s] = 3.0e38f; midx[s] = 0; }

  const int code_start = split * (KCODES / NSPLIT);
  const int nchunks = (KCODES / NSPLIT) / CHUNK;

  // Preload chunk 0 into buffer 0 via async DMA.
  {
    const float* g = w + (size_t)(code_start + r0) * CDIM + c4;
    unsigned l = lds_base0;
#pragma unroll
    for (int q = 0; q < 8; ++q) {
      async_copy16(l, g);
      g += 8 * CDIM;
      l += 8 * LDS_STRIDE * 4;
    }
  }
  wait_async0();
  __syncthreads();

  for (int ck = 0; ck < nchunks; ++ck) {
    const int cbase = code_start + ck * CHUNK;
    const int cur = ck & 1;

    // Kick off async staging of the next chunk into the other buffer.
    if (ck + 1 < nchunks) {
      const float* g = w + (size_t)(cbase + CHUNK + r0) * CDIM + c4;
      unsigned l = cur ? lds_base0 : lds_base1;
#pragma unroll
      for (int q = 0; q < 8; ++q) {
        async_copy16(l, g);
        g += 8 * CDIM;
        l += 8 * LDS_STRIDE * 4;
      }
    }

    const float* buf = lds[cur];
#pragma unroll
    for (int sub = 0; sub < CHUNK / 16; ++sub) {
      v8f acc = {0.f, 0.f, 0.f, 0.f, 0.f, 0.f, 0.f, 0.f};
      const float* bp = &buf[(sub * 16 + l15) * LDS_STRIDE + hi2];
#pragma unroll
      for (int kk = 0; kk < 32; ++kk) {
        const v2f b = *(const v2f*)(bp + kk * 4);
        acc = __builtin_amdgcn_wmma_f32_16x16x4_f32(
            /*neg_a=*/false, a[kk], /*neg_b=*/false, b,
            /*c_mod=*/(short)0, acc, /*reuse_a=*/false, /*reuse_b=*/false);
      }
      const int   code = cbase + sub * 16 + l15;
      const float h    = hn[code];
#pragma unroll
      for (int s = 0; s < 8; ++s) {
        const float sc = h - acc[s];      // 0.5||e||^2 - x.e
        if (sc < mval[s]) { mval[s] = sc; midx[s] = code; }
      }
    }

    wait_async0();       // own async DMAs into the other buffer are done
    __syncthreads();     // everyone's DMAs visible; all reads of cur finished
  }

  // Reduce across the 16 column-classes (lanes 0-15 / 16-31 are disjoint rows).
#pragma unroll
  for (int s = 0; s < 8; ++s) {
#pragma unroll
    for (int off = 1; off < 16; off <<= 1) {
      const float ov = __shfl_xor(mval[s], off, 32);
      const int   oi = __shfl_xor(midx[s], off, 32);
      if (ov < mval[s] || (ov == mval[s] && oi < midx[s])) { mval[s] = ov; midx[s] = oi; }
    }
  }
  if (l15 == 0) {
#pragma unroll
    for (int s = 0; s < 8; ++s) {
      const int row = row0 + s + (lane >> 4) * 8;   // D layout: VGPR s -> M=s / s+8
      pval[split * NROWS + row] = mval[s];
      pidx[split * NROWS + row] = midx[s];
    }
  }
}

// ---- kernel 3: reduce split partials to final indices ----------------------
__global__ __launch_bounds__(256) void finalize_kernel(const float* __restrict__ pval,
                                                       const int* __restrict__ pidx,
                                                       int* __restrict__ idx) {
  const int n = blockIdx.x * 256 + threadIdx.x;
  float bv = pval[n];
  int   bi = pidx[n];
#pragma unroll
  for (int s = 1; s < NSPLIT; ++s) {
    const float v = pval[s * NROWS + n];
    const int   i = pidx[s * NROWS + n];
    if (v < bv || (v == bv && i < bi)) { bv = v; bi = i; }
  }
  idx[n] = bi;
}

// ---- kernel 4: gather winning codes, NHWC -> NCHW --------------------------
__global__ __launch_bounds__(256) void gather_kernel(const float* __restrict__ w,
                                                     const int* __restrict__ idx,
                                                     float* __restrict__ outq) {
  const int o  = blockIdx.x * 256 + threadIdx.x;   // [8,128,32,32] linear
  const int b  = o >> 17;
  const int c  = (o >> 10) & 127;
  const int hw = o & 1023;
  const int n  = b * 1024 + hw;
  outq[o] = w[(size_t)idx[n] * CDIM + c];
}

// ---- kernel 5: fused zero-fill + one-hot scatter (256 MB nontemporal) ------
__global__ __launch_bounds__(256) void encodings_kernel(const int* __restrict__ idx,
                                                        v4f* __restrict__ enc) {
  const int total = (KCODES / 4) * NROWS;          // 16,777,216 float4
  for (int i = blockIdx.x * blockDim.x + threadIdx.x; i < total;
       i += gridDim.x * blockDim.x) {
    const int row = i >> 11;                       // 2048 float4 per row
    const int c0  = (i & 2047) * 4;
    const int t   = idx[row];
    v4f v;
    v.x = (t == c0    ) ? 1.f : 0.f;
    v.y = (t == c0 + 1) ? 1.f : 0.f;
    v.z = (t == c0 + 2) ? 1.f : 0.f;
    v.w = (t == c0 + 3) ? 1.f : 0.f;
    __builtin_nontemporal_store(v, enc + i);
  }
}

// ---------------------------------------------------------------------------
extern "C" void kernel_launch(void* const* d_in, const int* in_sizes, int n_in,
                              void* d_out, int out_size, void* d_ws, size_t ws_size,
                              hipStream_t stream) {
  const float* x = (const float*)d_in[0];          // [8,32,32,128] f32
  const float* w = (const float*)d_in[1];          // [8192,128]   f32

  float* outq = (float*)d_out;                     // quantized NCHW: 1,048,576 f32
  float* enc  = (float*)d_out + (8 * 128 * 32 * 32);  // one-hot: 67,108,864 f32

  char* ws = (char*)d_ws;
  float* hn   = (float*)ws;                                           // 32 KB
  float* pval = (float*)(ws + 32 * 1024);                             // 256 KB
  int*   pidx = (int*)(ws + 32 * 1024 + NSPLIT * NROWS * 4);          // 256 KB
  int*   idx  = (int*)(ws + 32 * 1024 + 2 * NSPLIT * NROWS * 4);      // 32 KB

  halfnorm_kernel<<<KCODES / 8, 256, 0, stream>>>(w, hn);
  argmin_kernel<<<(NROWS / ROWS_PER_WG) * NSPLIT, 256, 0, stream>>>(x, w, hn, pval, pidx);
  finalize_kernel<<<NROWS / 256, 256, 0, stream>>>(pval, pidx, idx);
  gather_kernel<<<(8 * 128 * 32 * 32) / 256, 256, 0, stream>>>(w, idx, outq);
  encodings_kernel<<<16384, 256, 0, stream>>>(idx, (v4f*)enc);
}